// DAGNNRecommender_6760278524490
// MI455X (gfx1250) — compile-verified
//
#include <hip/hip_runtime.h>
#include <hip/hip_bf16.h>
#include <math.h>

typedef float v2f __attribute__((ext_vector_type(2)));
typedef float v8f __attribute__((ext_vector_type(8)));

// ---------------------------------------------------------------------------
// GEMM: C[M,N] = A[M,K] @ B[K,N] + bias[N]  (fp32, V_WMMA_F32_16X16X4_F32)
// One wave computes a 16x64 tile of C (4 accumulators); the A-fragment is
// loaded once per K-step and reused by 4 WMMAs -> 4x less A traffic and
// 4x less address math per WMMA than a 16x16 tile.
// A-frag (16x4): lane lo = M row, VGPR v + 2*hi selects K.
// B-frag (4x16): row K = v + 2*hi, col = lane lo.
// C (16x16): row = v + 8*hi, col = lane lo.            (ISA 7.12.2)
// EXEC is all-1s at every WMMA: wave-uniform early-out only; per-lane bounds
// use clamped loads / guarded stores (no branches around the WMMAs).
// ---------------------------------------------------------------------------
__global__ __launch_bounds__(256) void gemm_bias_wmma(
    const float* __restrict__ A, const float* __restrict__ B,
    const float* __restrict__ bias, float* __restrict__ C,
    int M, int K, int N)
{
    const int lane = threadIdx.x & 31;
    const int wave = threadIdx.x >> 5;
    const int nTiles  = (N + 15) >> 4;
    const int nGroups = (nTiles + 3) >> 2;       // 4 N-tiles per wave
    const int mTiles  = (M + 15) >> 4;
    const int gw = blockIdx.x * 8 + wave;
    if (gw >= mTiles * nGroups) return;          // wave-uniform: EXEC stays full
    const int mt = gw / nGroups;
    const int ng = gw - mt * nGroups;
    const int row0 = mt << 4;
    const int hi = lane >> 4, lo = lane & 15;

    int arow = row0 + lo; if (arow > M - 1) arow = M - 1;    // clamp, no divergence
    const float* ap = A + (size_t)arow * K + (hi << 1);

    int col[4];
    const float* bp[4];
#pragma unroll
    for (int u = 0; u < 4; ++u) {
        col[u] = (ng << 6) + (u << 4) + lo;
        const int colc = (col[u] < N) ? col[u] : (N - 1);    // clamp, no divergence
        bp[u] = B + (size_t)(hi << 1) * N + colc;
    }

    v8f acc[4];
#pragma unroll
    for (int u = 0; u < 4; ++u) acc[u] = (v8f){0.f,0.f,0.f,0.f,0.f,0.f,0.f,0.f};

    for (int k0 = 0; k0 < K; k0 += 4) {
        v2f a; a.x = ap[k0]; a.y = ap[k0 + 1];               // one b64 load, reused 4x
#pragma unroll
        for (int u = 0; u < 4; ++u) {
            v2f b;
            b.x = bp[u][(size_t)k0 * N];
            b.y = bp[u][(size_t)k0 * N + N];
            acc[u] = __builtin_amdgcn_wmma_f32_16x16x4_f32(
                /*neg_a=*/false, a, /*neg_b=*/false, b,
                /*c_mod=*/(short)0, acc[u], /*reuse_a=*/false, /*reuse_b=*/false);
        }
    }

#pragma unroll
    for (int u = 0; u < 4; ++u) {
        if (col[u] < N) {
            const float bv = bias[col[u]];
#pragma unroll
            for (int v = 0; v < 8; ++v) {
                const int r = row0 + v + (hi << 3);
                if (r < M) C[(size_t)r * N + col[u]] = acc[u][v] + bv;
            }
        }
    }
}

// ---------------------------------------------------------------------------
// Utility / BN / DAGNN kernels
// ---------------------------------------------------------------------------
__global__ void fill_zero(float* __restrict__ p, int n) {
    int i = blockIdx.x * blockDim.x + threadIdx.x;
    if (i < n) p[i] = 0.f;
}

// per-column sum & sumsq over Nrows; blockDim.x == H; coalesced row-major reads
__global__ void colstats(const float* __restrict__ y, float* __restrict__ stats,
                         int Nrows, int H) {
    const int col = threadIdx.x;
    float s = 0.f, s2 = 0.f;
    for (int r = blockIdx.x; r < Nrows; r += gridDim.x) {
        float v = y[(size_t)r * H + col];
        s += v; s2 += v * v;
    }
    unsafeAtomicAdd(&stats[col], s);
    unsafeAtomicAdd(&stats[H + col], s2);
}

// (sum, sumsq) -> per-column affine (scale, shift) for fused BN apply
__global__ void finalize_stats(const float* __restrict__ stats,
                               const float* __restrict__ g, const float* __restrict__ be,
                               float* __restrict__ scl, float* __restrict__ shf,
                               float invN, int H) {
    int c = threadIdx.x;
    if (c >= H) return;
    float m   = stats[c] * invN;
    float var = stats[H + c] * invN - m * m;
    float s   = g[c] * rsqrtf(var + 1e-5f);
    scl[c] = s;
    shf[c] = be[c] - m * s;
}

__global__ void bn_relu_apply(float* __restrict__ y, const float* __restrict__ scl,
                              const float* __restrict__ shf, int total, int hmask) {
    int idx = blockIdx.x * blockDim.x + threadIdx.x;
    if (idx >= total) return;
    int c = idx & hmask;
    y[idx] = fmaxf(y[idx] * scl[c] + shf[c], 0.f);
}

// h2 = relu(bn(y2)) + h ; out = w[0]*h2 ; zero h-buffer (becomes first hop target)
__global__ void bn_relu_res_init(float* __restrict__ y2, float* h,
                                 float* __restrict__ outw,
                                 const float* __restrict__ scl, const float* __restrict__ shf,
                                 const float* __restrict__ w, int total, int hmask) {
    int idx = blockIdx.x * blockDim.x + threadIdx.x;
    if (idx >= total) return;
    int c = idx & hmask;
    float v = fmaxf(y2[idx] * scl[c] + shf[c], 0.f) + h[idx];
    y2[idx]   = v;
    outw[idx] = w[0] * v;
    h[idx]    = 0.f;
}

// softmax over the K+1 attention logits (single lane; n == 11)
__global__ void softmax_att(const float* __restrict__ att, float* __restrict__ w, int n) {
    if (threadIdx.x == 0 && blockIdx.x == 0) {
        float mx = att[0];
        for (int i = 1; i < n; ++i) mx = fmaxf(mx, att[i]);
        float s = 0.f;
        for (int i = 0; i < n; ++i) { float e = expf(att[i] - mx); w[i] = e; s += e; }
        float inv = 1.f / s;
        for (int i = 0; i < n; ++i) w[i] *= inv;
    }
}

// one hop: nxt[dst] += cur[src], H=128 hardcoded, float4 gather + 4 f32 atomics
__global__ void scatter_add_h128(const float* __restrict__ cur, float* __restrict__ nxt,
                                 const int* __restrict__ src, const int* __restrict__ dst,
                                 int E) {
    int idx = blockIdx.x * blockDim.x + threadIdx.x;
    if (idx >= (E << 5)) return;
    int e = idx >> 5;                 // one wave per edge (32 float4 quads)
    int j = idx & 31;
    int s = src[e], d = dst[e];
    const float4 v = ((const float4*)(cur + ((size_t)s << 7)))[j];
    float* dp = nxt + ((size_t)d << 7) + (j << 2);
    unsafeAtomicAdd(dp + 0, v.x);
    unsafeAtomicAdd(dp + 1, v.y);
    unsafeAtomicAdd(dp + 2, v.z);
    unsafeAtomicAdd(dp + 3, v.w);
}

// out += w[hop]*nxt ; zero the buffer just consumed (next hop's scatter target)
__global__ void axpy_zero(float4* __restrict__ out, const float4* __restrict__ nxt,
                          float4* __restrict__ zbuf, const float* __restrict__ w,
                          int hop, int n4) {
    int idx = blockIdx.x * blockDim.x + threadIdx.x;
    if (idx >= n4) return;
    float wv = w[hop];
    float4 o = out[idx];
    float4 v = nxt[idx];
    o.x += wv * v.x; o.y += wv * v.y; o.z += wv * v.z; o.w += wv * v.w;
    out[idx]  = o;
    zbuf[idx] = make_float4(0.f, 0.f, 0.f, 0.f);
}

// ---------------------------------------------------------------------------
extern "C" void kernel_launch(void* const* d_in, const int* in_sizes, int n_in,
                              void* d_out, int out_size, void* d_ws, size_t ws_size,
                              hipStream_t stream) {
    const float* x    = (const float*)d_in[0];
    const int*   ei   = (const int*)  d_in[1];
    const float* W1   = (const float*)d_in[2];
    const float* b1   = (const float*)d_in[3];
    const float* g1   = (const float*)d_in[4];
    const float* be1  = (const float*)d_in[5];
    const float* W2   = (const float*)d_in[6];
    const float* b2   = (const float*)d_in[7];
    const float* g2   = (const float*)d_in[8];
    const float* be2  = (const float*)d_in[9];
    const float* att  = (const float*)d_in[10];
    const float* W3   = (const float*)d_in[11];
    const float* b3   = (const float*)d_in[12];
    const float* g3   = (const float*)d_in[13];
    const float* be3  = (const float*)d_in[14];
    const float* Wout = (const float*)d_in[15];
    const float* bout = (const float*)d_in[16];

    const int H = 128, H2 = 64, OUTC = 1000, KHOPS = 10, KATT = 11;
    const int N = in_sizes[0] / H;   // 50000
    const int E = in_sizes[1] / 2;   // 800000
    const int* srcIdx = ei;
    const int* dstIdx = ei + E;

    float* bufA  = (float*)d_ws;                  // N*H   (h / hop buffer)
    float* bufB  = bufA + (size_t)N * H;          // N*H   (y2/h2 / hop buffer)
    float* bufO  = bufB + (size_t)N * H;          // N*H   (weighted hop accumulator)
    float* stats = bufO + (size_t)N * H;          // 2*H
    float* scl   = stats + 2 * H;                 // H
    float* shf   = scl + H;                       // H
    float* w     = shf + H;                       // KATT

    const int TB = 256;
    auto cdiv = [](long long a, long long b) { return (int)((a + b - 1) / b); };
    auto gemmBlocks = [&](int M_, int N_) {
        long long nGroups = cdiv(cdiv(N_, 16), 4);
        long long waves = (long long)cdiv(M_, 16) * nGroups;
        return cdiv(waves, 8);
    };

    // ---- layer 1: h = relu(bn(x@W1 + b1)) -> bufA ----
    gemm_bias_wmma<<<gemmBlocks(N, H), TB, 0, stream>>>(x, W1, b1, bufA, N, H, H);
    fill_zero<<<1, 2 * H, 0, stream>>>(stats, 2 * H);
    colstats<<<1024, H, 0, stream>>>(bufA, stats, N, H);
    finalize_stats<<<1, H, 0, stream>>>(stats, g1, be1, scl, shf, 1.0f / N, H);
    bn_relu_apply<<<cdiv((long long)N * H, TB), TB, 0, stream>>>(bufA, scl, shf, N * H, H - 1);

    // ---- layer 2: h2 = relu(bn(h@W2 + b2)) + h -> bufB ; out = w0*h2 ; zero bufA ----
    gemm_bias_wmma<<<gemmBlocks(N, H), TB, 0, stream>>>(bufA, W2, b2, bufB, N, H, H);
    fill_zero<<<1, 2 * H, 0, stream>>>(stats, 2 * H);
    colstats<<<1024, H, 0, stream>>>(bufB, stats, N, H);
    finalize_stats<<<1, H, 0, stream>>>(stats, g2, be2, scl, shf, 1.0f / N, H);
    softmax_att<<<1, 32, 0, stream>>>(att, w, KATT);
    bn_relu_res_init<<<cdiv((long long)N * H, TB), TB, 0, stream>>>(
        bufB, bufA, bufO, scl, shf, w, N * H, H - 1);

    // ---- DAGNN: K scatter-add hops, weighted accumulation into bufO ----
    for (int i = 0; i < KHOPS; ++i) {
        float* cur = (i & 1) ? bufA : bufB;
        float* nxt = (i & 1) ? bufB : bufA;
        scatter_add_h128<<<cdiv((long long)E * 32, TB), TB, 0, stream>>>(
            cur, nxt, srcIdx, dstIdx, E);
        axpy_zero<<<cdiv((long long)N * H / 4, TB), TB, 0, stream>>>(
            (float4*)bufO, (const float4*)nxt, (float4*)cur, w, i + 1, N * H / 4);
    }

    // ---- layer 3: h3 = relu(bn(bufO@W3 + b3)) -> bufA (free after hops) ----
    float* y3 = bufA;
    gemm_bias_wmma<<<gemmBlocks(N, H2), TB, 0, stream>>>(bufO, W3, b3, y3, N, H, H2);
    fill_zero<<<1, 2 * H2, 0, stream>>>(stats, 2 * H2);
    colstats<<<1024, H2, 0, stream>>>(y3, stats, N, H2);
    finalize_stats<<<1, H2, 0, stream>>>(stats, g3, be3, scl, shf, 1.0f / N, H2);
    bn_relu_apply<<<cdiv((long long)N * H2, TB), TB, 0, stream>>>(y3, scl, shf, N * H2, H2 - 1);

    // ---- output: d_out = h3 @ Wout + bout  [N, 1000] ----
    gemm_bias_wmma<<<gemmBlocks(N, OUTC), TB, 0, stream>>>(y3, Wout, bout, (float*)d_out, N, H2, OUTC);
}